// N3Tree_88184268521774
// MI455X (gfx1250) — compile-verified
//
#include <hip/hip_runtime.h>
#include <hip/hip_bf16.h>

// Octree vertical query, N=2, DEPTH=7, DATA_DIM=4.
// out[q] = sum over the 8 nodes on the root->leaf path of data[node, cx,cy,cz, :].
//
// MI455X (gfx1250, wave32, 23.3 TB/s HBM, 192 MB L2) design:
//  - Path bits extracted once: u = floor(coord * 2^8); bit (7-l) of u is the
//    cell coordinate at level l (exact binary bit peeling, matches the
//    reference's ind*2 - floor(ind*2) recurrence in fp32 exactly).
//  - Levels 0..2 collapsed into a 512-entry prefix table (node id + partial
//    float4 sum), built once by a 1-block kernel, staged per-block into LDS
//    (immune to WGP$ thrash from the 380 MB gather stream). Dependent chain
//    shrinks from 8 global loads to: 1 LDS + 4 global (child) + 1 leaf load.
//  - Leaf level (7): child delta == 0 by construction -> skip the child load
//    (saves 4M random 4B reads over a 67 MB array, i.e. full cachelines from
//    HBM). Leaf data load is non-temporal (268 MB region, ~2x reuse max) so
//    L2 keeps levels 0..6 (~58 MB) resident. Output stored non-temporally.
//  - 2 queries per thread, hand-interleaved: doubles outstanding loads per
//    wave (latency-BW product needs ~16 MB in flight to saturate HBM) and
//    halves per-query LDS staging cost.

typedef float v4f __attribute__((ext_vector_type(4)));

#define NTBL 512           // 8^3 path prefixes for levels 0..2
#define QBLOCK 256
#define QPT 2              // queries per thread (independent chains for MLP)

__device__ __forceinline__ int cell_of(int ux, int uy, int uz, int l) {
  const int s = 7 - l;
  return (((ux >> s) & 1) << 2) | (((uy >> s) & 1) << 1) | ((uz >> s) & 1);
}

__device__ __forceinline__ void coord_bits(const float* __restrict__ ip,
                                           int& ux, int& uy, int& uz) {
  float x = ip[0], y = ip[1], z = ip[2];
  ux = min(max((int)floorf(x * 256.0f), 0), 255);
  uy = min(max((int)floorf(y * 256.0f), 0), 255);
  uz = min(max((int)floorf(z * 256.0f), 0), 255);
}

__device__ __forceinline__ int prefix_of(int ux, int uy, int uz) {
  return (cell_of(ux, uy, uz, 0) << 6) |
         (cell_of(ux, uy, uz, 1) << 3) |
          cell_of(ux, uy, uz, 2);
}

// ---- Kernel 1: build 512-entry prefix table (levels 0..2) ------------------
__global__ void n3tree_build_table(const float* __restrict__ data,
                                   const int* __restrict__ child,
                                   int* __restrict__ tbl_node,
                                   float* __restrict__ tbl_acc) {
  const int t = threadIdx.x;            // 512 threads, one per 9-bit prefix
  const v4f* __restrict__ d4 = (const v4f*)data;
  int node = 0;
  v4f acc = (v4f){0.f, 0.f, 0.f, 0.f};
#pragma unroll
  for (int l = 0; l < 3; ++l) {
    const int cell = (t >> (6 - 3 * l)) & 7;
    const int off = node * 8 + cell;
    acc += d4[off];
    node += child[off];
  }
  tbl_node[t] = node;
  ((v4f*)tbl_acc)[t] = acc;
}

// ---- Kernel 2: dual-query traversal with LDS-staged prefix table -----------
__global__ void __launch_bounds__(QBLOCK)
n3tree_query(const float* __restrict__ data,
             const int* __restrict__ child,
             const float* __restrict__ indices,
             const int* __restrict__ tbl_node,
             const float* __restrict__ tbl_acc,
             float* __restrict__ out, int Q) {
  __shared__ int s_node[NTBL];
  __shared__ v4f s_acc[NTBL];
  for (int i = threadIdx.x; i < NTBL; i += QBLOCK) {
    s_node[i] = tbl_node[i];
    s_acc[i] = ((const v4f*)tbl_acc)[i];
  }
  __syncthreads();

  const int base = blockIdx.x * (QBLOCK * QPT) + threadIdx.x;
  const int qa = base;           // both coalesced across the wave
  const int qb = base + QBLOCK;
  const bool va = (qa < Q), vb = (qb < Q);

  // Out-of-range lanes traverse the (0,0,0) path (always in-bounds) and
  // simply skip the final store -> branchless interleaved traversal.
  int uxa = 0, uya = 0, uza = 0, uxb = 0, uyb = 0, uzb = 0;
  if (va) coord_bits(indices + 3 * (size_t)qa, uxa, uya, uza);
  if (vb) coord_bits(indices + 3 * (size_t)qb, uxb, uyb, uzb);

  const int pa = prefix_of(uxa, uya, uza);
  const int pb = prefix_of(uxb, uyb, uzb);

  int na = s_node[pa];
  int nb = s_node[pb];
  v4f aa = s_acc[pa];
  v4f ab = s_acc[pb];

  const v4f* __restrict__ d4 = (const v4f*)data;
  // Levels 3..6: L2-resident (~48 MB). Two independent chains interleaved so
  // each wave keeps 2x loads in flight; data loads overlap the child chain
  // via split load counters.
#pragma unroll
  for (int l = 3; l < 7; ++l) {
    const int offa = na * 8 + cell_of(uxa, uya, uza, l);
    const int offb = nb * 8 + cell_of(uxb, uyb, uzb, l);
    const v4f va4 = d4[offa];
    const int da = child[offa];
    const v4f vb4 = d4[offb];
    const int db = child[offb];
    aa += va4; na += da;
    ab += vb4; nb += db;
  }
  // Level 7 (leaf): delta == 0 by construction -> no child load; NT data load.
  const int off7a = na * 8 + cell_of(uxa, uya, uza, 7);
  const int off7b = nb * 8 + cell_of(uxb, uyb, uzb, 7);
  aa += __builtin_nontemporal_load(d4 + off7a);
  ab += __builtin_nontemporal_load(d4 + off7b);

  if (va) __builtin_nontemporal_store(aa, (v4f*)out + qa);
  if (vb) __builtin_nontemporal_store(ab, (v4f*)out + qb);
}

// ---- Fallback (no workspace): full 8-level traversal -----------------------
__global__ void __launch_bounds__(QBLOCK)
n3tree_query_notable(const float* __restrict__ data,
                     const int* __restrict__ child,
                     const float* __restrict__ indices,
                     float* __restrict__ out, int Q) {
  const int q = blockIdx.x * blockDim.x + threadIdx.x;
  if (q >= Q) return;

  int ux, uy, uz;
  coord_bits(indices + 3 * (size_t)q, ux, uy, uz);

  const v4f* __restrict__ d4 = (const v4f*)data;
  int node = 0;
  v4f acc = (v4f){0.f, 0.f, 0.f, 0.f};
#pragma unroll
  for (int l = 0; l < 7; ++l) {
    const int off = node * 8 + cell_of(ux, uy, uz, l);
    const v4f v = d4[off];
    const int d = child[off];
    acc += v;
    node += d;
  }
  const int off7 = node * 8 + cell_of(ux, uy, uz, 7);
  acc += __builtin_nontemporal_load(d4 + off7);

  __builtin_nontemporal_store(acc, (v4f*)out + q);
}

extern "C" void kernel_launch(void* const* d_in, const int* in_sizes, int n_in,
                              void* d_out, int out_size, void* d_ws, size_t ws_size,
                              hipStream_t stream) {
  const float* data    = (const float*)d_in[0];   // (M,2,2,2,4) f32
  const int*   child   = (const int*)d_in[1];     // (M,2,2,2) i32
  const float* indices = (const float*)d_in[2];   // (Q,3) f32
  float* out = (float*)d_out;                     // (Q,4) f32

  const int Q = in_sizes[2] / 3;

  // Workspace layout: [0,8192) = 512 x float4 partial sums (16B aligned),
  //                   [8192,10240) = 512 x int node ids.
  if (ws_size >= 10240 && d_ws != nullptr) {
    float* tbl_acc = (float*)d_ws;
    int*   tbl_node = (int*)((char*)d_ws + 8192);
    const int blocks = (Q + QBLOCK * QPT - 1) / (QBLOCK * QPT);
    n3tree_build_table<<<1, NTBL, 0, stream>>>(data, child, tbl_node, tbl_acc);
    n3tree_query<<<blocks, QBLOCK, 0, stream>>>(data, child, indices,
                                                tbl_node, tbl_acc, out, Q);
  } else {
    const int blocks = (Q + QBLOCK - 1) / QBLOCK;
    n3tree_query_notable<<<blocks, QBLOCK, 0, stream>>>(data, child, indices,
                                                        out, Q);
  }
}